// BlockRevert_64553358459201
// MI455X (gfx1250) — compile-verified
//
#include <hip/hip_runtime.h>

// ---------------------------------------------------------------------------
// BlockRevert: out[b,s,0,:]   = tb[b,s,0,:]                + pe[s,:] + mod[0,:]
//              out[b,s,m,:]   = (idx<8 ? tb[b,s,1+idx,:]
//                                      : mask_token[:])    + pe[s,:] + mod[m,:]
// B=8 S=512 Mv=8 D=512 R=16  -> memory-bound (~293MB moved, ~13us @ 23.3TB/s)
// CDNA5 path: TDM (tensor_load_to_lds) stages mod_emb + mask_token into LDS,
// s_wait_tensorcnt syncs, main loop is pure coalesced b128 streaming.
// ---------------------------------------------------------------------------

typedef unsigned int u32x4 __attribute__((ext_vector_type(4)));
typedef int          i32x8 __attribute__((ext_vector_type(8)));
typedef int          i32x4 __attribute__((ext_vector_type(4)));

#define B_  8
#define S_  512
#define MV  8
#define D_  512
#define R_  16
#define M_  (R_ + 1)   // 17 output rows per (b,s)
#define D4  (D_ / 4)   // 128 float4 per row

// --- TDM descriptor group 0: count=1, lds_addr, 57-bit global addr, type=2 ---
__device__ __forceinline__ u32x4 tdm_g0(unsigned lds_addr, const void* gptr) {
    unsigned long long ga = (unsigned long long)(uintptr_t)gptr;
    u32x4 g;
    g.x = 1u;                                                // count=1, user mode
    g.y = lds_addr;                                          // LDS byte address
    g.z = (unsigned)ga;                                      // global_addr[31:0]
    g.w = ((unsigned)(ga >> 32) & 0x01FFFFFFu) | (2u << 30); // addr[56:32] | type=2
    return g;
}

// --- TDM descriptor group 1: 2-D tensor, data_size=4B, contiguous rows ---
__device__ __forceinline__ i32x8 tdm_g1_2d(unsigned dim0, unsigned dim1,
                                           unsigned tile0, unsigned tile1,
                                           unsigned long long stride0) {
    i32x8 g;
    g[0] = (int)(2u << 16);                                  // data_size = 4 bytes
    g[1] = (int)((dim0 & 0xFFFFu) << 16);                    // tensor_dim0[15:0] @ bits63:48
    g[2] = (int)((dim0 >> 16) | ((dim1 & 0xFFFFu) << 16));   // dim0 hi | dim1 lo
    g[3] = (int)((dim1 >> 16) | (tile0 << 16));              // dim1 hi | tile_dim0
    g[4] = (int)(tile1 & 0xFFFFu);                           // tile_dim1 | tile_dim2=0
    g[5] = (int)(unsigned)(stride0 & 0xFFFFFFFFull);         // tensor_dim0_stride lo32
    g[6] = (int)(unsigned)((stride0 >> 32) & 0xFFFFull);     // stride hi16 | dim1_stride lo=0
    g[7] = 0;
    return g;
}

__global__ __launch_bounds__(256)
void block_revert_kernel(const float* __restrict__ tb,        // (B,S,9,D)
                         const float* __restrict__ mask_tok,  // (D)
                         const float* __restrict__ mod_emb,   // (17,D)
                         const int*   __restrict__ ridx,      // (B,S,R)
                         float*       __restrict__ out) {     // (B,S,17,D)
    __shared__ float smem[M_ * D_ + D_ + D_]; // [comb 17*512][mask 512][pe 512]
    __shared__ int   sidx[R_];
    float* s_comb = smem;                 // mod_emb, later mod+pe
    float* s_mask = smem + M_ * D_;       // mask_token
    float* s_pe   = smem + M_ * D_ + D_;  // sinusoidal PE row for this s

    const int bs  = blockIdx.x;           // 0..4095
    const int s   = bs & (S_ - 1);
    const int tid = threadIdx.x;

    // --- wave 0: TDM-stage mod_emb (17x512 f32) and mask_token (1x512) ---
    if (tid < 32) {
        unsigned lds_base = (unsigned)(unsigned long long)(uintptr_t)smem;
        i32x4 z4 = {0, 0, 0, 0};                    // groups 2/3 unused (2-D)
        i32x8 z8 = {0, 0, 0, 0, 0, 0, 0, 0};        // extra group (6-arg form)
        u32x4 g0a = tdm_g0(lds_base, mod_emb);
        i32x8 g1a = tdm_g1_2d(D_, M_, D_, M_, D_);
        __builtin_amdgcn_tensor_load_to_lds(g0a, g1a, z4, z4, z8, 0);
        u32x4 g0b = tdm_g0(lds_base + (unsigned)(M_ * D_ * 4), mask_tok);
        i32x8 g1b = tdm_g1_2d(D_, 1, D_, 1, D_);
        __builtin_amdgcn_tensor_load_to_lds(g0b, g1b, z4, z4, z8, 0);
        __builtin_amdgcn_s_wait_tensorcnt(0);
    }

    // --- concurrently: PE row for this s (one sin/cos pair per thread) ---
    {
        const float k   = -9.2103403719761836f / (float)D_;  // -ln(10000)/D
        float       div = __expf((float)(2 * tid) * k);
        float       ang = (float)s * div;
        s_pe[2 * tid]     = __sinf(ang);
        s_pe[2 * tid + 1] = __cosf(ang);
    }
    if (tid < R_) sidx[tid] = ridx[bs * R_ + tid];
    __syncthreads();

    // --- fuse: comb[m,d] = mod[m,d] + pe[s,d] ---
    for (int t = tid; t < M_ * D_; t += 256)
        s_comb[t] += s_pe[t & (D_ - 1)];
    __syncthreads();

    // --- stream output: b128 loads/stores, wave-uniform row index ---
    const float4* tb4   = (const float4*)(tb + (size_t)bs * (1 + MV) * D_);
    const float4* mask4 = (const float4*)s_mask;
    const float4* comb4 = (const float4*)s_comb;
    float4*       out4  = (float4*)(out + (size_t)bs * M_ * D_);

    for (int t = tid; t < M_ * D4; t += 256) {
        int m = t >> 7;          // row 0..16 (uniform per wave: 128 % 32 == 0)
        int q = t & (D4 - 1);    // float4 column
        float4 a = comb4[m * D4 + q];
        float4 v;
        if (m == 0) {
            v = tb4[q];                                   // global token row
        } else {
            int idx = sidx[m - 1];                        // wave-uniform
            v = (idx < MV) ? tb4[(1 + idx) * D4 + q]      // valid modality row
                           : mask4[q];                    // mask token (LDS)
        }
        float4 r;
        r.x = a.x + v.x; r.y = a.y + v.y; r.z = a.z + v.z; r.w = a.w + v.w;
        out4[m * D4 + q] = r;
    }
}

extern "C" void kernel_launch(void* const* d_in, const int* in_sizes, int n_in,
                              void* d_out, int out_size, void* d_ws, size_t ws_size,
                              hipStream_t stream) {
    (void)in_sizes; (void)n_in; (void)d_ws; (void)ws_size; (void)out_size;
    const float* tb       = (const float*)d_in[0]; // (8,512,9,512) f32
    const float* mask_tok = (const float*)d_in[1]; // (512) f32
    const float* mod_emb  = (const float*)d_in[2]; // (17,512) f32
    const int*   ridx     = (const int*)d_in[3];   // (8,512,16) int
    float*       out      = (float*)d_out;         // (8,512,17,512) f32

    dim3 grid(B_ * S_);   // 4096 workgroups, one per (b,s)
    dim3 block(256);      // 8 waves (wave32)
    block_revert_kernel<<<grid, block, 0, stream>>>(tb, mask_tok, mod_emb, ridx, out);
}